// GridSamplePScan_39874476376599
// MI455X (gfx1250) — compile-verified
//
#include <hip/hip_runtime.h>
#include <math.h>

// ---- problem dims (from reference) ----
#define Bdim 2
#define Ldim 48
#define Cdim 8
#define Hdim 48
#define Wdim 48
#define HW   (Hdim * Wdim)        // 2304
#define CPB  2                    // channels per block
#define CSPLIT (Cdim / CPB)       // 4
#define NTHREADS 256
#define PXT (HW / NTHREADS)       // 9 pixels per thread
#define TILE_ELTS  (CPB * HW)     // 4608 f32
#define TILE_BYTES (TILE_ELTS * 4) // 18432 B per LDS buffer

typedef unsigned int u32x4 __attribute__((ext_vector_type(4)));
typedef int          i32x8 __attribute__((ext_vector_type(8)));
typedef int          i32x4 __attribute__((ext_vector_type(4)));

// Issue a TDM DMA: contiguous TILE_BYTES from global -> LDS (1-D tensor descriptor).
// D# layout per CDNA5 ISA ch.8 (group0: count/lds_addr/global_addr/type,
// group1: data_size / tensor_dim0 / tensor_dim1=1 / tile_dim0 / dim0_stride).
// 6-arg builtin form (clang-23 toolchain): groups 0..3 + extra group + cpol.
__device__ __forceinline__ void tdm_issue(const float* gsrc, unsigned lds_off) {
  unsigned long long ga = (unsigned long long)(uintptr_t)gsrc;
  u32x4 g0;
  g0.x = 1u;                                               // count=1, user descriptor
  g0.y = lds_off;                                          // lds_addr (bytes)
  g0.z = (unsigned)(ga & 0xFFFFFFFFull);                   // global_addr[31:0]
  g0.w = (unsigned)((ga >> 32) & 0x01FFFFFFull)            // global_addr[56:32]
         | (2u << 30);                                     // type = 2 ("image")
  i32x8 g1;
  g1[0] = (2 << 16);                                       // data_size = 4 bytes
  g1[1] = (int)((TILE_ELTS & 0xFFFF) << 16);               // tensor_dim0[15:0]
  g1[2] = (int)(((TILE_ELTS >> 16) & 0xFFFF) | (1 << 16)); // td0[31:16] | tensor_dim1=1
  g1[3] = (int)((TILE_ELTS & 0xFFFF) << 16);               // td1[31:16]=0 | tile_dim0
  g1[4] = 0;                                               // tile_dim1=0, tile_dim2=0
  g1[5] = TILE_ELTS;                                       // tensor_dim0_stride[31:0]
  g1[6] = 0;                                               // stride hi | dim1_stride lo
  g1[7] = 0;                                               // dim1_stride hi
  i32x4 gz4 = {0, 0, 0, 0};                                // groups 2/3 unused (<=2D)
  i32x8 gz8 = {0, 0, 0, 0, 0, 0, 0, 0};                    // trailing group (unused)
  __builtin_amdgcn_tensor_load_to_lds(g0, g1, gz4, gz4, gz8, 0);
}

__global__ void __launch_bounds__(NTHREADS)
GridSamplePScan_kernel(const float* __restrict__ flows,   // (B,L,2,H,W)
                       const float* __restrict__ images,  // (B,L,C,H,W)
                       float* __restrict__ out)           // (B,L,C,H,W)
{
  extern __shared__ float lbuf[];   // 2 * TILE_ELTS floats (double buffer)

  const int tid  = threadIdx.x;
  const int bt   = blockIdx.x / CSPLIT;
  const int csel = blockIdx.x % CSPLIT;
  const int b    = bt / Ldim;
  const int t    = bt % Ldim;
  const int c0   = csel * CPB;

  float acc[PXT][CPB];
  float relx[PXT], rely[PXT];
  float bx[PXT], by[PXT];          // (gx+1), (gy+1) at zero flow

  #pragma unroll
  for (int i = 0; i < PXT; ++i) {
    relx[i] = 0.f; rely[i] = 0.f;
    #pragma unroll
    for (int c = 0; c < CPB; ++c) acc[i][c] = 0.f;
    int p = tid + i * NTHREADS;
    int h = p / Wdim, w = p % Wdim;
    bx[i] = (float)(2 * w + 1) * (1.0f / (float)Wdim);
    by[i] = (float)(2 * h + 1) * (1.0f / (float)Hdim);
  }

  const unsigned lds_base = (unsigned)(uintptr_t)(&lbuf[0]); // LDS byte offset

  // Prologue: stage k = t into buffer 0 (wave 0 only; TDM descriptors are uniform).
  if (tid < 32) {
    tdm_issue(images + ((((size_t)b * Ldim + t) * Cdim + c0) * HW), lds_base);
  }

  int cur = 0;
  for (int k = t; k >= 0; --k) {
    if (tid < 32) {
      if (k > 0) {
        // Prefetch next slab into the other buffer, then wait for the current
        // one (TENSORcnt is in-order per wave: <=1 means the older DMA is done).
        tdm_issue(images + ((((size_t)b * Ldim + (k - 1)) * Cdim + c0) * HW),
                  lds_base + (unsigned)((cur ^ 1) * TILE_BYTES));
        __builtin_amdgcn_s_wait_tensorcnt(1);
      } else {
        __builtin_amdgcn_s_wait_tensorcnt(0);
      }
    }
    __syncthreads();   // slab `cur` visible to all waves

    const float* __restrict__ src   = &lbuf[cur * TILE_ELTS];
    const float* __restrict__ flowp = flows + (((size_t)b * Ldim + k) * 2) * HW;

    #pragma unroll
    for (int i = 0; i < PXT; ++i) {
      const int p = tid + i * NTHREADS;

      // grid coords: gx wraps with remainder(gx+1, 2); gy is plain.
      float u  = bx[i] + relx[i];                    // gx + 1
      float r  = u - 2.0f * floorf(0.5f * u);        // remainder(u, 2) in [0,2)
      float ix = r * (0.5f * (float)Wdim) - 0.5f;
      float iy = (by[i] + rely[i]) * (0.5f * (float)Hdim) - 0.5f;

      float x0f = floorf(ix), y0f = floorf(iy);
      float fx = ix - x0f, fy = iy - y0f;
      int   x0 = (int)x0f,  y0 = (int)y0f;

      float vx0 = (x0f >= 0.f        && x0f <= (float)(Wdim - 1)) ? 1.f : 0.f;
      float vx1 = (x0f + 1.f >= 0.f  && x0f + 1.f <= (float)(Wdim - 1)) ? 1.f : 0.f;
      float vy0 = (y0f >= 0.f        && y0f <= (float)(Hdim - 1)) ? 1.f : 0.f;
      float vy1 = (y0f + 1.f >= 0.f  && y0f + 1.f <= (float)(Hdim - 1)) ? 1.f : 0.f;

      int x0c = min(max(x0, 0), Wdim - 1);
      int x1c = min(max(x0 + 1, 0), Wdim - 1);
      int y0c = min(max(y0, 0), Hdim - 1);
      int y1c = min(max(y0 + 1, 0), Hdim - 1);

      float w00 = (1.f - fx) * (1.f - fy) * vx0 * vy0;
      float w10 = fx         * (1.f - fy) * vx1 * vy0;
      float w01 = (1.f - fx) * fy         * vx0 * vy1;
      float w11 = fx         * fy         * vx1 * vy1;

      int i00 = y0c * Wdim + x0c;
      int i10 = y0c * Wdim + x1c;
      int i01 = y1c * Wdim + x0c;
      int i11 = y1c * Wdim + x1c;

      #pragma unroll
      for (int c = 0; c < CPB; ++c) {
        const float* sc = src + c * HW;
        acc[i][c] = fmaf(w00, sc[i00],
                    fmaf(w10, sc[i10],
                    fmaf(w01, sc[i01],
                    fmaf(w11, sc[i11], acc[i][c]))));
      }

      // advance running relative flow toward k-1: rel += flow[k]
      relx[i] += flowp[p];
      rely[i] += flowp[HW + p];
    }
    __syncthreads();   // everyone done reading `cur` before it is overwritten
    cur ^= 1;
  }

  float* op = out + (((size_t)b * Ldim + t) * Cdim + c0) * HW;
  #pragma unroll
  for (int i = 0; i < PXT; ++i) {
    const int p = tid + i * NTHREADS;
    #pragma unroll
    for (int c = 0; c < CPB; ++c) op[c * HW + p] = acc[i][c];
  }
}

extern "C" void kernel_launch(void* const* d_in, const int* in_sizes, int n_in,
                              void* d_out, int out_size, void* d_ws, size_t ws_size,
                              hipStream_t stream) {
  (void)in_sizes; (void)n_in; (void)out_size; (void)d_ws; (void)ws_size;
  const float* flows  = (const float*)d_in[0];   // (B,L,2,H,W) f32
  const float* images = (const float*)d_in[1];   // (B,L,C,H,W) f32
  float*       out    = (float*)d_out;           // (B,L,C,H,W) f32

  dim3 grid(Bdim * Ldim * CSPLIT);               // 384 blocks
  dim3 block(NTHREADS);                          // 8 wave32s
  size_t shmem = (size_t)(2 * TILE_BYTES);       // 36 KB double buffer
  GridSamplePScan_kernel<<<grid, block, shmem, stream>>>(flows, images, out);
}